// EnergyNet_11699490915124
// MI455X (gfx1250) — compile-verified
//
#include <hip/hip_runtime.h>
#include <hip/hip_bf16.h>
#include <math.h>

// MI455X / gfx1250, wave32. Compile-only loop: aim for clean codegen with
// v_wmma_f32_16x16x4_f32 in both the per-atom feature pack and the pairwise tile.

#define NN 2048
#define CONV 332.07156f

typedef float v2f __attribute__((ext_vector_type(2)));
typedef float v8f __attribute__((ext_vector_type(8)));

// ---------------------------------------------------------------------------
// Kernel 1: per-atom feature pack.
// feat[n][i], n = 0:a(sf_lo) 1:b(sf_hi) 2:rlo 3:rhi 4:dlo 5:dhi 6:die 7:born
// Computed as D(16x16) = embs(16x8) x G(8x16) via two fp32 WMMA K-steps.
// One wave (32 threads) per 16-atom tile.
// ---------------------------------------------------------------------------
__device__ __forceinline__ float g_coef(int k, int n,
                                        const float* __restrict__ sf,
                                        const float* __restrict__ born,
                                        const float* __restrict__ die,
                                        const float* __restrict__ depth,
                                        const float* __restrict__ radius) {
  switch (n) {
    case 0: return sf[k];
    case 1: return sf[8 + k];
    case 2: return radius[k];
    case 3: return radius[8 + k];
    case 4: return depth[k];
    case 5: return depth[8 + k];
    case 6: return die[k];
    case 7: return born[k];
    default: return 0.0f;
  }
}

__global__ __launch_bounds__(32) void feat_kernel(
    const float* __restrict__ embs,
    const float* __restrict__ sf_elec,
    const float* __restrict__ born,
    const float* __restrict__ die,
    const float* __restrict__ depth,
    const float* __restrict__ radius,
    float* __restrict__ feat) {
  const int lane = threadIdx.x & 31;
  const int n = lane & 15;          // feature column
  const int half = lane >> 4;
  const int base = blockIdx.x * 16; // atom tile

  const int m = base + (lane & 15); // A-matrix row for this lane

  // A layout (16x4 f32): lanes 0-15 -> K={0,1}, lanes 16-31 -> K={2,3}
  v2f a0, a1, b0, b1;
  a0.x = embs[m * 8 + 2 * half];
  a0.y = embs[m * 8 + 2 * half + 1];
  a1.x = embs[m * 8 + 4 + 2 * half];
  a1.y = embs[m * 8 + 5 + 2 * half];
  // B layout (4x16 f32), mirrored: lanes 0-15 -> K={0,1}, lanes 16-31 -> K={2,3}
  b0.x = g_coef(2 * half,     n, sf_elec, born, die, depth, radius);
  b0.y = g_coef(2 * half + 1, n, sf_elec, born, die, depth, radius);
  b1.x = g_coef(4 + 2 * half, n, sf_elec, born, die, depth, radius);
  b1.y = g_coef(5 + 2 * half, n, sf_elec, born, die, depth, radius);

  v8f c = {0.f, 0.f, 0.f, 0.f, 0.f, 0.f, 0.f, 0.f};
  c = __builtin_amdgcn_wmma_f32_16x16x4_f32(false, a0, false, b0, (short)0, c, false, false);
  c = __builtin_amdgcn_wmma_f32_16x16x4_f32(false, a1, false, b1, (short)0, c, false, false);

  if (n < 8) {
#pragma unroll
    for (int r = 0; r < 8; ++r) {
      // C/D layout: VGPR r, lanes 0-15 -> row r, lanes 16-31 -> row r+8
      feat[n * NN + base + r + 8 * half] = c[r];
    }
  }
}

// ---------------------------------------------------------------------------
// Kernel 2: pairwise 16x16 tiles. Each wave owns one tile in the WMMA C-layout;
// OPP = Ps_i . Ps_j for the tile is one v_wmma_f32_16x16x4_f32.
// Grid: (128 i-tiles, 4 j-slices), 256 threads (8 waves) per block.
// Deterministic per-block partial sums written to workspace.
// ---------------------------------------------------------------------------
__global__ __launch_bounds__(256) void pair_kernel(
    const float* __restrict__ X,
    const float* __restrict__ qs,
    const float* __restrict__ Ps,
    const float* __restrict__ w0,
    const float* __restrict__ s0,
    const int* __restrict__ chain,
    const float* __restrict__ feat,
    const float* __restrict__ sf_elec,
    float* __restrict__ partials) {
  const int lane = threadIdx.x & 31;
  const int wave = threadIdx.x >> 5;
  const int col = lane & 15;
  const int half = lane >> 4;
  const int ibase = blockIdx.x * 16;
  const float c_sf = sf_elec[16];

  // Hoist row (i) data: 8 rows per lane (rows half*8 .. half*8+7 of the tile).
  float rXx[8], rXy[8], rXz[8], rq[8], rb[8], rrhi[8], rdhi[8];
  float rPx[8], rPy[8], rPz[8];
  int rch[8];
#pragma unroll
  for (int r = 0; r < 8; ++r) {
    const int i = ibase + r + 8 * half;
    rXx[r] = X[i * 3 + 0];
    rXy[r] = X[i * 3 + 1];
    rXz[r] = X[i * 3 + 2];
    rq[r] = qs[i];
    rch[r] = chain[i];
    rb[r] = feat[1 * NN + i];
    rrhi[r] = feat[3 * NN + i];
    rdhi[r] = feat[5 * NN + i];
    rPx[r] = Ps[i * 3 + 0];
    rPy[r] = Ps[i * 3 + 1];
    rPz[r] = Ps[i * 3 + 2];
  }

  // A-matrix = Ps rows of this i-tile (16x4, K=3 padded with 0).
  const int ia = ibase + col;
  v2f A;
  A.x = half ? Ps[ia * 3 + 2] : Ps[ia * 3 + 0];
  A.y = half ? 0.0f : Ps[ia * 3 + 1];

  float acc_e = 0.0f, acc_v = 0.0f;

  for (int jt = blockIdx.y * 8 + wave; jt < 128; jt += 32) {
    const int j = jt * 16 + col;
    const float Xjx = X[j * 3 + 0], Xjy = X[j * 3 + 1], Xjz = X[j * 3 + 2];
    const float qj = qs[j];
    const int chj = chain[j];
    const float s0j = s0[j];
    const float w0v = w0[j];
    const float w0j = sqrtf(w0v * w0v + 1e-6f);
    const float aj = feat[0 * NN + j];
    const float rloj = feat[2 * NN + j];
    const float dloj = feat[4 * NN + j];

    // B-matrix = Ps rows of the j-tile (columns of Ps^T).
    v2f B;
    B.x = half ? Ps[j * 3 + 2] : Ps[j * 3 + 0];
    B.y = half ? 0.0f : Ps[j * 3 + 1];

    v8f opp = {0.f, 0.f, 0.f, 0.f, 0.f, 0.f, 0.f, 0.f};
    opp = __builtin_amdgcn_wmma_f32_16x16x4_f32(false, A, false, B, (short)0, opp, false, false);

#pragma unroll
    for (int r = 0; r < 8; ++r) {
      const float dx = Xjx - rXx[r];
      const float dy = Xjy - rXy[r];
      const float dz = Xjz - rXz[r];
      const float r2 = dx * dx + dy * dy + dz * dz;
      const float D = sqrtf(r2 + 3e-6f);           // sum(V*V + 1e-6) over 3 comps
      const float invD = 1.0f / (D + 1e-6f);
      const float mask = (chj != rch[r]) ? 1.0f : 0.0f;

      // --- electrostatics ---
      const float vpx = dx + 1e-6f, vpy = dy + 1e-6f, vpz = dz + 1e-6f;
      float nn = sqrtf(vpx * vpx + vpy * vpy + vpz * vpz);
      nn = fmaxf(nn, 1e-12f);
      const float oqp = (vpx * rPx[r] + vpy * rPy[r] + vpz * rPz[r]) / nn;
      const float sf = aj + rb[r] + invD * c_sf;
      const float invD2 = invD * invD;
      const float invD4 = invD2 * invD2;
      const float invD6 = invD4 * invD2;
      const float eterm = rq[r] * qj * invD * sf + oqp * invD4 + opp[r] * invD6;
      acc_e += mask * eterm;

      // --- van der Waals ---
      const float sig_r = 1.0f / (1.0f + __expf(-(rloj + rrhi[r])));
      const float s = 2.0f * s0j * (0.8f * sig_r + 0.4f);
      const float repl = 5.0f * __expf(-0.3f * D * D * D);
      const float Dm = D - s;
      const float Dm2 = Dm * Dm;
      const float dsh = Dm - 0.3f;
      const float attr = (__expf(-dsh * dsh) + __expf(-3.0f * Dm2) +
                          __expf(-10.0f * Dm2)) * (1.0f / 3.0f);
      const float sig_d = 1.0f / (1.0f + __expf(-(dloj + rdhi[r])));
      const float w = w0j * (sig_d + 0.5f);
      acc_v += mask * (repl - w * attr);
    }
  }

  // Deterministic block reduction.
  __shared__ float se[256], sv[256];
  se[threadIdx.x] = acc_e;
  sv[threadIdx.x] = acc_v;
  __syncthreads();
  for (int sft = 128; sft > 0; sft >>= 1) {
    if (threadIdx.x < (unsigned)sft) {
      se[threadIdx.x] += se[threadIdx.x + sft];
      sv[threadIdx.x] += sv[threadIdx.x + sft];
    }
    __syncthreads();
  }
  if (threadIdx.x == 0) {
    const int bid = blockIdx.y * gridDim.x + blockIdx.x;
    partials[bid * 2 + 0] = se[0];
    partials[bid * 2 + 1] = sv[0];
  }
}

// ---------------------------------------------------------------------------
// Kernel 3: finalize — sum block partials (fixed order), compute E_solv,
// apply scales + NaN guards, write (E_vdw, E_elec, E_solv).
// ---------------------------------------------------------------------------
__global__ __launch_bounds__(256) void finalize_kernel(
    const float* __restrict__ partials, int n_part,
    const float* __restrict__ feat,
    const float* __restrict__ qs,
    float* __restrict__ out) {
  const int t = threadIdx.x;
  float es = 0.0f, ev = 0.0f, sv = 0.0f;
  for (int k = t; k < n_part; k += 256) {
    es += partials[2 * k + 0];
    ev += partials[2 * k + 1];
  }
  for (int i = t; i < NN; i += 256) {
    const float die = feat[6 * NN + i] + 1e-6f;
    const float R = feat[7 * NN + i] + 1.0f;
    sv += -(1.0f - 1.0f / die) * qs[i] / (R + 1e-6f);
  }
  __shared__ float a0[256], a1[256], a2[256];
  a0[t] = es; a1[t] = ev; a2[t] = sv;
  __syncthreads();
  for (int sft = 128; sft > 0; sft >>= 1) {
    if (t < (unsigned)sft) {
      a0[t] += a0[t + sft];
      a1[t] += a1[t + sft];
      a2[t] += a2[t + sft];
    }
    __syncthreads();
  }
  if (t == 0) {
    float E_elec = 0.5f * CONV * a0[0];
    float E_vdw = a1[0];
    float E_solv = CONV * a2[0] * 0.01f;
    if (E_elec != E_elec) E_elec = 1e-6f;
    if (E_vdw != E_vdw) E_vdw = 1e-6f;
    if (E_solv != E_solv) E_solv = 1e-6f;
    out[0] = E_vdw;   // reference returns (E_vdw, E_elec, E_solv)
    out[1] = E_elec;
    out[2] = E_solv;
  }
}

// ---------------------------------------------------------------------------
extern "C" void kernel_launch(void* const* d_in, const int* in_sizes, int n_in,
                              void* d_out, int out_size, void* d_ws, size_t ws_size,
                              hipStream_t stream) {
  (void)in_sizes; (void)n_in; (void)out_size; (void)ws_size;
  const float* X       = (const float*)d_in[0];
  const float* embs    = (const float*)d_in[1];
  const float* qs      = (const float*)d_in[2];
  const float* Ps      = (const float*)d_in[3];
  const float* w0      = (const float*)d_in[4];
  const float* s0      = (const float*)d_in[5];
  const int*   chain   = (const int*)d_in[6];
  const float* sf_elec = (const float*)d_in[7];
  const float* born    = (const float*)d_in[8];
  const float* die     = (const float*)d_in[9];
  const float* depth   = (const float*)d_in[10];
  const float* radius  = (const float*)d_in[11];
  float* out = (float*)d_out;

  float* feat = (float*)d_ws;            // 8 * 2048 floats (64 KB)
  float* partials = feat + 8 * NN;       // 512 blocks * 2 floats

  feat_kernel<<<NN / 16, 32, 0, stream>>>(embs, sf_elec, born, die, depth, radius, feat);
  pair_kernel<<<dim3(NN / 16, 4), 256, 0, stream>>>(X, qs, Ps, w0, s0, chain, feat,
                                                    sf_elec, partials);
  finalize_kernel<<<1, 256, 0, stream>>>(partials, (NN / 16) * 4, feat, qs, out);
}